// GCNLayer_56719338111684
// MI455X (gfx1250) — compile-verified
//
#include <hip/hip_runtime.h>
#include <hip/hip_bf16.h>

// ---------------------------------------------------------------------------
// GCN layer for MI455X (gfx1250):
//   deg-norm scatter-add (L2-resident, wave-per-edge, f32 atomics, prefetch)
//   -> f32 WMMA GEMM (V_WMMA_F32_16X16X4_F32, 16x16 tiles, LDS-staged A)
//   -> fused column-stat accumulation -> BN + ReLU in place.
// ---------------------------------------------------------------------------

typedef float v2f __attribute__((ext_vector_type(2)));
typedef float v8f __attribute__((ext_vector_type(8)));

#define D_FEAT 128
#define BN_EPS 1e-5f

// ---- kernel 1: zero scratch region (grid-stride) ---------------------------
__global__ __launch_bounds__(256) void gcn_zero_kernel(float* p, int count) {
    int i = blockIdx.x * blockDim.x + threadIdx.x;
    int stride = gridDim.x * blockDim.x;
    for (; i < count; i += stride) p[i] = 0.0f;
}

// ---- kernel 2: degree histograms ------------------------------------------
__global__ __launch_bounds__(256) void gcn_degree_kernel(const int* __restrict__ src,
                                                         const int* __restrict__ dst,
                                                         float* deg_out, float* deg_in,
                                                         int E) {
    int i = blockIdx.x * blockDim.x + threadIdx.x;
    int stride = gridDim.x * blockDim.x;
    for (; i < E; i += stride) {
        atomicAdd(&deg_out[src[i]], 1.0f);
        atomicAdd(&deg_in[dst[i]], 1.0f);
    }
}

// ---- kernel 3: edge aggregation  agg[dst] += x[src] * rsqrt(deg_out[src]) --
// One wave32 per edge; each lane owns 4 contiguous channels (float4 load).
__global__ __launch_bounds__(256) void gcn_aggregate_kernel(const float* __restrict__ x,
                                                            const float* __restrict__ deg_out,
                                                            const int* __restrict__ src,
                                                            const int* __restrict__ dst,
                                                            float* agg, int E) {
    int wave   = (blockIdx.x * blockDim.x + threadIdx.x) >> 5;
    int lane   = threadIdx.x & 31;
    int nwaves = (gridDim.x * blockDim.x) >> 5;
    for (int e = wave; e < E; e += nwaves) {
        int s = src[e];
        int d = dst[e];
        // prefetch next edge's source row into the cache hierarchy
        int en = e + nwaves;
        if (en < E) {
            int sn = src[en];
            __builtin_prefetch(&x[(size_t)sn * D_FEAT + lane * 4], 0, 3);
        }
        float scale = rsqrtf(fmaxf(deg_out[s], 1.0f));
        const float4 xv = *(const float4*)(&x[(size_t)s * D_FEAT + lane * 4]);
        float* ap = &agg[(size_t)d * D_FEAT + lane * 4];
        atomicAdd(ap + 0, xv.x * scale);
        atomicAdd(ap + 1, xv.y * scale);
        atomicAdd(ap + 2, xv.z * scale);
        atomicAdd(ap + 3, xv.w * scale);
    }
}

// ---- kernel 4: WMMA GEMM + bias + column-stat accumulation ----------------
// Block = 256 threads = 8 wave32s. Block handles a 16-row strip of A (scaled
// by rsqrt(deg_in)); wave w computes the 16x16 output tile at columns
// [16w, 16w+16). K-loop: 32 x V_WMMA_F32_16X16X4_F32.
//
// f32 WMMA VGPR layouts (ISA 7.12.2):
//   A 16x4 : lane l<16 -> row M=l, holds K={k0, k0+1}; lanes 16-31 -> K={k0+2, k0+3}
//   B 4x16 : lane l<16 -> col N=l, holds K={k0, k0+1}; lanes 16-31 -> K={k0+2, k0+3}
//   C 16x16: vgpr r, lanes 0-15 -> (M=r, N=lane); lanes 16-31 -> (M=r+8, N=lane-16)
__global__ __launch_bounds__(256) void gcn_gemm_wmma_kernel(const float* __restrict__ agg,
                                                            const float* __restrict__ deg_in,
                                                            const float* __restrict__ W,
                                                            const float* __restrict__ bias,
                                                            float* __restrict__ out,
                                                            float* colsum, float* colsumsq,
                                                            int N) {
    __shared__ float rowscale[16];
    __shared__ float Atile[16 * 132];   // 128-col rows padded to 132 -> conflict-free

    const int tid  = threadIdx.x;
    const int row0 = blockIdx.x * 16;

    if (tid < 16) {
        int row = row0 + tid;
        rowscale[tid] = (row < N) ? rsqrtf(fmaxf(deg_in[row], 1.0f)) : 0.0f;
    }
    __syncthreads();

    // stage the 16x128 A tile (scaled) into LDS
    for (int i = tid; i < 16 * D_FEAT; i += 256) {
        int r = i >> 7;
        int k = i & 127;
        int row = row0 + r;
        float v = 0.0f;
        if (row < N) v = agg[(size_t)row * D_FEAT + k] * rowscale[r];
        Atile[r * 132 + k] = v;
    }
    __syncthreads();

    const int wave = tid >> 5;
    const int lane = tid & 31;
    const int half = lane >> 4;     // which K pair this lane supplies
    const int l16  = lane & 15;
    const int col  = wave * 16 + l16;

    v8f c = {};
#pragma unroll
    for (int k0 = 0; k0 < D_FEAT; k0 += 4) {
        const int kk = k0 + 2 * half;
        v2f a, b;
        a.x = Atile[l16 * 132 + kk];
        a.y = Atile[l16 * 132 + kk + 1];
        b.x = W[(size_t)kk * D_FEAT + col];
        b.y = W[(size_t)(kk + 1) * D_FEAT + col];
        // 8 args: (neg_a, A, neg_b, B, c_mod, C, reuse_a, reuse_b)
        c = __builtin_amdgcn_wmma_f32_16x16x4_f32(false, a, false, b,
                                                  (short)0, c, false, false);
    }

    // epilogue: bias, store pre-BN output, accumulate column stats
    const float bv = bias[col];
    float psum = 0.0f, psq = 0.0f;
#pragma unroll
    for (int r = 0; r < 8; ++r) {
        int row = row0 + half * 8 + r;
        if (row < N) {
            float v = c[r] + bv;
            out[(size_t)row * D_FEAT + col] = v;
            psum += v;
            psq  += v * v;
        }
    }
    atomicAdd(&colsum[col], psum);
    atomicAdd(&colsumsq[col], psq);
}

// ---- kernel 5: finalize BN statistics -------------------------------------
__global__ void gcn_stats_kernel(const float* colsum, const float* colsumsq,
                                 float* mean, float* invstd, int N) {
    int c = threadIdx.x;
    float invN = 1.0f / (float)N;
    float m = colsum[c] * invN;
    float v = colsumsq[c] * invN - m * m;
    mean[c]   = m;
    invstd[c] = rsqrtf(fmaxf(v, 0.0f) + BN_EPS);
}

// ---- kernel 6: BN + ReLU in place -----------------------------------------
__global__ __launch_bounds__(256) void gcn_bn_relu_kernel(float* out,
                                                          const float* __restrict__ mean,
                                                          const float* __restrict__ invstd,
                                                          const float* __restrict__ gamma,
                                                          const float* __restrict__ beta,
                                                          int total) {
    int i = blockIdx.x * blockDim.x + threadIdx.x;
    int stride = gridDim.x * blockDim.x;
    for (; i < total; i += stride) {
        int ch = i & 127;
        float v = (out[i] - mean[ch]) * invstd[ch] * gamma[ch] + beta[ch];
        out[i] = fmaxf(v, 0.0f);
    }
}

// ---------------------------------------------------------------------------
extern "C" void kernel_launch(void* const* d_in, const int* in_sizes, int n_in,
                              void* d_out, int out_size, void* d_ws, size_t ws_size,
                              hipStream_t stream) {
    const float* x     = (const float*)d_in[0];
    const float* W     = (const float*)d_in[1];
    const float* bias  = (const float*)d_in[2];
    const float* gamma = (const float*)d_in[3];
    const float* beta  = (const float*)d_in[4];
    const int*   src   = (const int*)d_in[5];
    const int*   dst   = (const int*)d_in[6];

    const int N = in_sizes[0] / D_FEAT;
    const int E = in_sizes[5];

    // workspace layout (floats): deg_out[N] deg_in[N] agg[N*128]
    //                            colsum[128] colsumsq[128] mean[128] invstd[128]
    float* ws       = (float*)d_ws;
    float* deg_out  = ws;
    float* deg_in   = deg_out + N;
    float* agg      = deg_in + N;
    float* colsum   = agg + (size_t)N * D_FEAT;
    float* colsumsq = colsum + D_FEAT;
    float* mean     = colsumsq + D_FEAT;
    float* invstd   = mean + D_FEAT;

    float* out = (float*)d_out;

    // 1. zero degrees, agg, col stats (contiguous prefix of ws)
    const int zero_count = 2 * N + N * D_FEAT + 2 * D_FEAT;
    gcn_zero_kernel<<<4096, 256, 0, stream>>>(ws, zero_count);

    // 2. degrees
    gcn_degree_kernel<<<(E + 255) / 256, 256, 0, stream>>>(src, dst, deg_out, deg_in, E);

    // 3. edge aggregation (wave per edge)
    gcn_aggregate_kernel<<<4096, 256, 0, stream>>>(x, deg_out, src, dst, agg, E);

    // 4. WMMA GEMM + bias + column stats
    gcn_gemm_wmma_kernel<<<(N + 15) / 16, 256, 0, stream>>>(agg, deg_in, W, bias, out,
                                                            colsum, colsumsq, N);

    // 5. finalize mean / invstd
    gcn_stats_kernel<<<1, D_FEAT, 0, stream>>>(colsum, colsumsq, mean, invstd, N);

    // 6. BN + ReLU in place
    gcn_bn_relu_kernel<<<4096, 256, 0, stream>>>(out, mean, invstd, gamma, beta, N * D_FEAT);
}